// RGCN_6416681141169
// MI455X (gfx1250) — compile-verified
//
#include <hip/hip_runtime.h>

typedef __attribute__((ext_vector_type(2))) float v2f;
typedef __attribute__((ext_vector_type(8))) float v8f;

#define N_NODES 50000
#define N_EDGES 640000
#define N_REL   8
#define KDIM    128   // IN_DIM == HID_DIM == 128
#define NSEG    (N_REL + 1)

// ---------------------------------------------------------------------------
// Zero the per-relation aggregation buffer.
// ---------------------------------------------------------------------------
__global__ void rgcn_zero_kernel(float4* __restrict__ p, long long n4) {
    long long i = blockIdx.x * (long long)blockDim.x + threadIdx.x;
    if (i < n4) p[i] = make_float4(0.f, 0.f, 0.f, 0.f);
}

// ---------------------------------------------------------------------------
// Scatter: aggR[etype[e]][dst[e]][o] += feat[src[e]][o]
// One thread per (edge, channel); coalesced row reads, coalesced f32 atomics
// (resolved at L2).
// ---------------------------------------------------------------------------
__global__ void rgcn_scatter_kernel(const float* __restrict__ feat,   // [N][128]
                                    const int*   __restrict__ esrc,
                                    const int*   __restrict__ edst,
                                    const int*   __restrict__ etyp,
                                    float*       __restrict__ aggR,   // [8][N][128]
                                    int nEdges) {
    long long tid = blockIdx.x * (long long)blockDim.x + threadIdx.x;
    int e = (int)(tid >> 7);
    int o = (int)(tid & 127);
    if (e >= nEdges) return;
    int s = esrc[e];
    int d = edst[e];
    int r = etyp[e];
    float v = feat[(size_t)s * KDIM + o];
    atomicAdd(aggR + ((size_t)r * N_NODES + (size_t)d) * KDIM + o, v);
}

// ---------------------------------------------------------------------------
// Pack weights into per-wave WMMA B-fragment order so the GEMM inner loop
// needs exactly one global_load_b64 (base + immediate offset) per K-step.
//
//   Wf[((seg*tiles + t)*32 + k4)*32 + lane] = { W[k0][n], W[k0+1][n] }
//   k0 = k4*4 + (lane>>4)*2,  n = t*16 + (lane&15)
//   W  = seg<8 ? Wrel + seg*K*O : Wself
//
// Matches CDNA5 ISA 7.12.2 32-bit B 4x16 layout (lanes 0-15: K pair {0,1},
// lanes 16-31: K pair {2,3}; N = lane&15).
// ---------------------------------------------------------------------------
__global__ void rgcn_pack_w_kernel(const float* __restrict__ Wrel,   // [8][128][O]
                                   const float* __restrict__ Wself,  // [128][O]
                                   float2*      __restrict__ Wf,
                                   int O) {
    const int tiles = O >> 4;
    const int total = NSEG * tiles * 32 * 32;
    int idx = blockIdx.x * blockDim.x + threadIdx.x;
    if (idx >= total) return;
    int lane = idx & 31;
    int k4   = (idx >> 5) & 31;
    int t    = (idx >> 10) % tiles;
    int seg  = idx / (tiles << 10);
    const float* W = (seg < N_REL) ? (Wrel + (size_t)seg * KDIM * O) : Wself;
    int k0 = k4 * 4 + (lane >> 4) * 2;
    int n  = t * 16 + (lane & 15);
    Wf[idx] = make_float2(W[(size_t)k0 * O + n], W[(size_t)(k0 + 1) * O + n]);
}

// ---------------------------------------------------------------------------
// Fused RGCN transform:
//   out[n][o] = act( sum_{r} aggR[r][n][:] @ Wrel[r][:][o]
//                    + hself[n][:] @ Wself[:][o] + bias[o] )
//
// Block = 16 rows x O cols; one wave32 per 16x16 tile. A tile staged in LDS
// (row stride 132 -> conflict-free ds_load_2addr_b64 fragment reads). B read
// from the pre-swizzled fragment buffer: one b64 load with immediate offset
// per WMMA — no address math in the hot loop.
// ---------------------------------------------------------------------------
__global__ void rgcn_gemm_kernel(const float*  __restrict__ aggR,   // [8][N][128]
                                 const float*  __restrict__ hself,  // [N][128]
                                 const float2* __restrict__ Wf,     // packed frags
                                 const float*  __restrict__ bias,   // [O]
                                 float*        __restrict__ out,    // [N][O]
                                 int O, int applyRelu) {
    __shared__ float lds[16 * 132];

    const int m0    = blockIdx.x * 16;
    const int lane  = threadIdx.x & 31;
    const int wave  = threadIdx.x >> 5;
    const int half  = lane >> 4;
    const int l15   = lane & 15;
    const int nb    = wave * 16;
    const int tiles = O >> 4;

    v8f c = {0.f, 0.f, 0.f, 0.f, 0.f, 0.f, 0.f, 0.f};

    for (int s = 0; s < NSEG; ++s) {
        const float*  A  = (s < N_REL) ? (aggR + (size_t)s * N_NODES * KDIM) : hself;
        // per-wave fragment base: 32 k-steps x 32 lanes of float2
        const float2* wf = Wf + ((size_t)(s * tiles + wave) << 10) + lane;

        __syncthreads();
        for (int t = threadIdx.x; t < 16 * KDIM; t += blockDim.x) {
            int row = t >> 7;
            int col = t & 127;
            lds[row * 132 + col] = A[(size_t)(m0 + row) * KDIM + col];
        }
        __syncthreads();

#pragma unroll
        for (int k4 = 0; k4 < 32; ++k4) {
            const float* ap = &lds[l15 * 132 + k4 * 4 + half * 2];
            v2f a;
            a.x = ap[0];
            a.y = ap[1];
            float2 bf = wf[k4 * 32];          // global_load_b64, imm offset
            v2f b;
            b.x = bf.x;
            b.y = bf.y;
            c = __builtin_amdgcn_wmma_f32_16x16x4_f32(
                    false, a, false, b, (short)0, c, false, false);
        }
    }

    float bv = bias[nb + l15];
#pragma unroll
    for (int v = 0; v < 8; ++v) {
        float x = c[v] + bv;
        if (applyRelu) x = fmaxf(x, 0.f);
        out[(size_t)(m0 + v + 8 * half) * O + nb + l15] = x;
    }
}

// ---------------------------------------------------------------------------
// Orchestration. ws layout (floats):
//   aggR [8][50000][128]          51,200,000   (reused by both layers)
//   h1   [50000][128]              6,400,000
//   Wf1  9*8*32*32 float2            147,456
//   Wf2  9*4*32*32 float2             73,728
// Total ~231.3 MB.
// ---------------------------------------------------------------------------
extern "C" void kernel_launch(void* const* d_in, const int* in_sizes, int n_in,
                              void* d_out, int out_size, void* d_ws, size_t ws_size,
                              hipStream_t stream) {
    const float* features = (const float*)d_in[0];
    const int*   src      = (const int*)  d_in[1];
    const int*   dst      = (const int*)  d_in[2];
    const int*   etype    = (const int*)  d_in[3];
    const float* W1       = (const float*)d_in[4];
    const float* W1s      = (const float*)d_in[5];
    const float* b1       = (const float*)d_in[6];
    const float* W2       = (const float*)d_in[7];
    const float* W2s      = (const float*)d_in[8];
    const float* b2       = (const float*)d_in[9];
    float*       out      = (float*)d_out;

    float*  aggR = (float*)d_ws;
    float*  h1   = aggR + (size_t)N_REL * N_NODES * KDIM;
    float2* Wf1  = (float2*)(h1 + (size_t)N_NODES * KDIM);
    float2* Wf2  = Wf1 + (size_t)NSEG * 8 * 32 * 32;

    const long long n4     = (long long)N_REL * N_NODES * KDIM / 4; // 12.8M
    const int zeroBlocks   = (int)((n4 + 255) / 256);
    const int scatBlocks   = (int)(((long long)N_EDGES * KDIM) / 256); // 320000
    const int gemmBlocks   = N_NODES / 16;                             // 3125

    const int pack1 = NSEG * 8 * 32 * 32;   // 73728 fragment elements
    const int pack2 = NSEG * 4 * 32 * 32;   // 36864

    // Weight fragment packing (tiny, L2-resident)
    rgcn_pack_w_kernel<<<(pack1 + 255) / 256, 256, 0, stream>>>(W1, W1s, Wf1, 128);
    rgcn_pack_w_kernel<<<(pack2 + 255) / 256, 256, 0, stream>>>(W2, W2s, Wf2, 64);

    // ---- Layer 1: h1 = relu(sum_r aggR[r]@W1[r] + x@W1_self + b1) ----
    rgcn_zero_kernel<<<zeroBlocks, 256, 0, stream>>>((float4*)aggR, n4);
    rgcn_scatter_kernel<<<scatBlocks, 256, 0, stream>>>(
        features, src, dst, etype, aggR, N_EDGES);
    rgcn_gemm_kernel<<<gemmBlocks, 256, 0, stream>>>(   // O=128 -> 8 waves
        aggR, features, Wf1, b1, h1, 128, 1);

    // ---- Layer 2: out = sum_r aggR[r]@W2[r] + h1@W2_self + b2 ----
    rgcn_zero_kernel<<<zeroBlocks, 256, 0, stream>>>((float4*)aggR, n4);
    rgcn_scatter_kernel<<<scatBlocks, 256, 0, stream>>>(
        h1, src, dst, etype, aggR, N_EDGES);
    rgcn_gemm_kernel<<<gemmBlocks, 128, 0, stream>>>(   // O=64 -> 4 waves
        aggR, h1, Wf2, b2, out, 64, 0);
}